// MultiHeadAttention_79405355369100
// MI455X (gfx1250) — compile-verified
//
#include <hip/hip_runtime.h>

// MI455X / gfx1250, wave32. fp32 WMMA path: V_WMMA_F32_16X16X4_F32.
// A 16x4 frag: 2 VGPRs/lane (lanes 0-15: M=lane,K=0..1; lanes 16-31: M=lane-16,K=2..3)
// B 4x16 frag: 2 VGPRs/lane (lanes 0-15: N=lane,K=0/1; lanes 16-31: N=lane-16,K=2/3)
// C/D 16x16: 8 VGPRs (vgpr i: lanes 0-15 -> M=i, lanes 16-31 -> M=i+8; N=lane&15)
typedef float v2f __attribute__((ext_vector_type(2)));
typedef float v8f __attribute__((ext_vector_type(8)));

#define D_MODEL 1024
#define D_K     64
#define N_HEADS 16
#define BATCH   2
#define SEQ     2048
#define M_TOTAL (BATCH * SEQ) // 4096

// ---------------------------------------------------------------------------
// Tensor Data Mover (TDM) 2D tile load: global -> LDS, descriptor-driven.
// D# per CDNA5 ISA (group0 128b + group1 256b).
// Row padding: pad_interval code 4 => pad after 32 DWORDs, pad_amount code 0
// => +1 DWORD, i.e. exactly our float[rows][33] LDS layout.
// ---------------------------------------------------------------------------
#if __has_builtin(__builtin_amdgcn_tensor_load_to_lds) && \
    __has_builtin(__builtin_amdgcn_s_wait_tensorcnt)
#define HAVE_TDM 1
typedef unsigned int u32x4 __attribute__((ext_vector_type(4)));
typedef int i32x4 __attribute__((ext_vector_type(4)));
typedef int i32x8 __attribute__((ext_vector_type(8)));

__device__ __forceinline__ void tdm_load_tile_2d(
    unsigned lds_off, const void* gptr,
    unsigned tensor_d0, unsigned tensor_d1,      // tensor extents (elems)
    unsigned tile_d0, unsigned tile_d1,          // tile extents (elems)
    unsigned stride0_elems,                      // elems between rows
    unsigned pad_interval_code, unsigned pad_amount_code)
{
  unsigned long long ga = (unsigned long long)gptr;
  u32x4 g0;
  g0[0] = 1u;                                        // count=1 (valid user D#)
  g0[1] = lds_off;                                   // lds_addr (bytes)
  g0[2] = (unsigned)(ga & 0xffffffffu);              // global_addr[31:0]
  g0[3] = (unsigned)((ga >> 32) & 0x01ffffffu)       // global_addr[56:32]
          | 0x80000000u;                             // type=2 ("image")
  i32x8 g1;
  g1[0] = (int)((2u << 16)                           // data_size: 4 bytes
                | (1u << 20)                         // pad_enable
                | (pad_interval_code << 22)
                | (pad_amount_code << 25));
  g1[1] = (int)((tensor_d0 & 0xffffu) << 16);        // tensor_dim0[15:0]
  g1[2] = (int)((tensor_d0 >> 16) | ((tensor_d1 & 0xffffu) << 16));
  g1[3] = (int)((tensor_d1 >> 16) | (tile_d0 << 16)); // tile_dim0
  g1[4] = (int)tile_d1;                               // tile_dim1 (tile_dim2=0)
  g1[5] = (int)stride0_elems;                         // tensor_dim0_stride[31:0]
  g1[6] = 0;
  g1[7] = 0;
  i32x4 gz = {0, 0, 0, 0};
#if __clang_major__ >= 23
  i32x8 gz8 = {0, 0, 0, 0, 0, 0, 0, 0};
  __builtin_amdgcn_tensor_load_to_lds(g0, g1, gz, gz, gz8, 0);
#else
  __builtin_amdgcn_tensor_load_to_lds(g0, g1, gz, gz, 0);
#endif
}
#endif

// ---------------------------------------------------------------------------
// Generic GEMM: C[M,N] = A[M,K] @ W[K,N] + bias[N] (+ residual[M,N])
// 256 threads = 8 waves; workgroup tile 64(M) x 32(N); waves 4x2, 16x16 each.
// K staged through LDS in 32-wide slabs; tiles staged by the TDM (wave 0
// issues the DMA descriptors, waits TENSORcnt, barrier releases the pack).
// ---------------------------------------------------------------------------
__global__ __launch_bounds__(256)
void gemm_wmma_f32(const float* __restrict__ A, const float* __restrict__ W,
                   const float* __restrict__ bias, const float* __restrict__ residual,
                   float* __restrict__ C, int M, int N, int K)
{
  __shared__ float lA[64][33];
  __shared__ float lB[32][33];

  const int tid  = threadIdx.x;
  const int lane = tid & 31;
  const int wid  = tid >> 5;   // 0..7
  const int wm   = wid >> 1;   // 0..3 (M sub-tile)
  const int wn   = wid & 1;    // 0..1 (N sub-tile)
  const int m0   = blockIdx.x * 64;
  const int n0   = blockIdx.y * 32;

  const int r  = lane & 15;    // A row / B col within 16x16 tile
  const int hi = lane >> 4;    // 0 or 1
  const int ko = hi * 2;       // K sub-offset for A/B fragments

  v8f acc = {};

  for (int kt = 0; kt < K; kt += 32) {
#ifdef HAVE_TDM
    if (wid == 0) {  // wave-uniform: one wave drives both tile DMAs
      tdm_load_tile_2d((unsigned)(unsigned long long)&lA[0][0],
                       &A[(size_t)m0 * K + kt], (unsigned)K, (unsigned)M,
                       32u, 64u, (unsigned)K, 4u, 0u);
      tdm_load_tile_2d((unsigned)(unsigned long long)&lB[0][0],
                       &W[(size_t)kt * N + n0], (unsigned)N, (unsigned)K,
                       32u, 32u, (unsigned)N, 4u, 0u);
      __builtin_amdgcn_s_wait_tensorcnt(0);
    }
#else
    for (int i = tid; i < 64 * 32; i += 256) {
      int rr = i >> 5, cc = i & 31;
      lA[rr][cc] = A[(size_t)(m0 + rr) * K + kt + cc];
    }
    for (int i = tid; i < 32 * 32; i += 256) {
      int rr = i >> 5, cc = i & 31;
      lB[rr][cc] = W[(size_t)(kt + rr) * N + n0 + cc];
    }
#endif
    __syncthreads();

    const float* aRow = &lA[wm * 16 + r][0];
#pragma unroll
    for (int kk = 0; kk < 32; kk += 4) {
      v2f a, b;
      a.x = aRow[kk + ko];
      a.y = aRow[kk + ko + 1];
      b.x = lB[kk + ko][wn * 16 + r];
      b.y = lB[kk + ko + 1][wn * 16 + r];
      acc = __builtin_amdgcn_wmma_f32_16x16x4_f32(false, a, false, b,
                                                  (short)0, acc, false, false);
    }
    __syncthreads();
  }

#pragma unroll
  for (int i = 0; i < 8; i++) {
    int gm = m0 + wm * 16 + i + hi * 8;
    int gn = n0 + wn * 16 + r;
    float v = acc[i] + bias[gn];
    if (residual) v += residual[(size_t)gm * N + gn];
    C[(size_t)gm * N + gn] = v;
  }
}

// ---------------------------------------------------------------------------
// Fused attention: one workgroup per (b, h, 16-row q block).
// Scores (WMMA, q from LDS, kT streamed from L2 w/ prefetch) -> mask -> LDS
// row buffer (16 x 2048 fp32 = 128KB < 320KB/WGP) -> wave32 shuffle softmax
// (attn written once, non-temporal) -> context = P @ V (WMMA from LDS).
// ---------------------------------------------------------------------------
__global__ __launch_bounds__(256)
void attn_wmma(const float* __restrict__ qp, const float* __restrict__ kp,
               const float* __restrict__ vp, const unsigned char* __restrict__ em,
               float* __restrict__ attn_out, float* __restrict__ ctx)
{
  extern __shared__ float smem[];
  float* sQ = smem;            // 16 x 64  (pre-scaled by 1/sqrt(d_k))
  float* sS = smem + 16 * 64;  // 16 x 2048 score/prob buffer

  const int tid  = threadIdx.x;
  const int lane = tid & 31;
  const int wid  = tid >> 5;
  const int q0   = blockIdx.x * 16;
  const int h    = blockIdx.y;
  const int b    = blockIdx.z;

  const int r  = lane & 15;
  const int hi = lane >> 4;
  const int ko = hi * 2;

  for (int i = tid; i < 16 * 64; i += 256) {
    int rr = i >> 6, cc = i & 63;
    sQ[i] = qp[(size_t)(b * SEQ + q0 + rr) * D_MODEL + h * D_K + cc] * 0.125f;
  }
  __syncthreads();

  // ---- scores: 128 column tiles of 16, 16 tiles per wave ----
  for (int j = wid; j < SEQ / 16; j += 8) {
    v8f acc = {};
    const float* aRow  = sQ + r * 64;
    const float* kBase = kp + (size_t)(b * SEQ + j * 16 + r) * D_MODEL + h * D_K;
    // prefetch this lane's k-row for the next tile this wave will process
    if (j + 8 < SEQ / 16)
      __builtin_prefetch(kBase + (size_t)8 * 16 * D_MODEL, 0, 1);
#pragma unroll
    for (int kk = 0; kk < 64; kk += 4) {
      v2f a, bf;
      a.x  = aRow[kk + ko];   a.y  = aRow[kk + ko + 1];
      bf.x = kBase[kk + ko];  bf.y = kBase[kk + ko + 1];
      acc = __builtin_amdgcn_wmma_f32_16x16x4_f32(false, a, false, bf,
                                                  (short)0, acc, false, false);
    }
#pragma unroll
    for (int i = 0; i < 8; i++) {
      int row  = i + hi * 8;
      int colg = j * 16 + r;
      unsigned char m = em[(size_t)(b * SEQ + q0 + row) * SEQ + colg];
      sS[row * SEQ + colg] = m ? -1e9f : acc[i];
    }
  }
  __syncthreads();

  // ---- softmax: each wave owns 2 rows, wave32 shuffle reductions ----
  for (int rr = wid * 2; rr < wid * 2 + 2; rr++) {
    float* row = sS + rr * SEQ;
    float mx = -3.4e38f;
    for (int c = lane; c < SEQ; c += 32) mx = fmaxf(mx, row[c]);
    for (int o = 16; o > 0; o >>= 1) mx = fmaxf(mx, __shfl_xor(mx, o, 32));
    float sum = 0.f;
    for (int c = lane; c < SEQ; c += 32) {
      float e = __expf(row[c] - mx);
      row[c] = e;
      sum += e;
    }
    for (int o = 16; o > 0; o >>= 1) sum += __shfl_xor(sum, o, 32);
    float inv = 1.f / sum;
    float* ao = attn_out + ((size_t)(b * N_HEADS + h) * SEQ + (q0 + rr)) * SEQ;
    for (int c = lane; c < SEQ; c += 32) {
      float p = row[c] * inv;
      row[c] = p;
      // attn tensor: 537MB write-once stream -> non-temporal, keep L2 for k/v
      __builtin_nontemporal_store(p, &ao[c]);
    }
  }
  __syncthreads();

  // ---- context = P(16x2048) @ V(2048x64): 4 waves, one 16x16 N-tile each ----
  if (wid < 4) {
    const int n0w = wid * 16;
    v8f acc = {};
    for (int k0 = 0; k0 < SEQ; k0 += 4) {
      v2f a, bf;
      const float* aRow = sS + r * SEQ + k0 + ko;
      a.x = aRow[0];
      a.y = aRow[1];
      const float* vB = vp + (size_t)(b * SEQ + k0 + ko) * D_MODEL + h * D_K + n0w + r;
      bf.x = vB[0];
      bf.y = vB[D_MODEL];
      acc = __builtin_amdgcn_wmma_f32_16x16x4_f32(false, a, false, bf,
                                                  (short)0, acc, false, false);
    }
#pragma unroll
    for (int i = 0; i < 8; i++) {
      int row = i + hi * 8;
      ctx[(size_t)(b * SEQ + q0 + row) * D_MODEL + h * D_K + n0w + r] = acc[i];
    }
  }
}

// ---------------------------------------------------------------------------
// LayerNorm over D_MODEL per row: out = (x - mu) * rsqrt(var + eps) * g + b
// ---------------------------------------------------------------------------
__global__ __launch_bounds__(256)
void layernorm_k(const float* __restrict__ x, const float* __restrict__ gamma,
                 const float* __restrict__ beta, float* __restrict__ out)
{
  __shared__ float red0[8];
  __shared__ float red1[8];
  const int row = blockIdx.x;
  const int tid = threadIdx.x;
  const float* xr = x + (size_t)row * D_MODEL;

  float s = 0.f, s2 = 0.f;
  for (int c = tid; c < D_MODEL; c += 256) {
    float v = xr[c];
    s += v;
    s2 += v * v;
  }
  for (int o = 16; o > 0; o >>= 1) {
    s  += __shfl_xor(s, o, 32);
    s2 += __shfl_xor(s2, o, 32);
  }
  if ((tid & 31) == 0) { red0[tid >> 5] = s; red1[tid >> 5] = s2; }
  __syncthreads();
  float ts = 0.f, ts2 = 0.f;
#pragma unroll
  for (int i = 0; i < 8; i++) { ts += red0[i]; ts2 += red1[i]; }
  float mu  = ts * (1.f / D_MODEL);
  float var = ts2 * (1.f / D_MODEL) - mu * mu;
  float rs  = rsqrtf(var + 1e-5f);
  float* orow = out + (size_t)row * D_MODEL;
  for (int c = tid; c < D_MODEL; c += 256) {
    float v = (xr[c] - mu) * rs * gamma[c] + beta[c];
    __builtin_nontemporal_store(v, &orow[c]);  // final output: streaming write
  }
}

// ---------------------------------------------------------------------------
extern "C" void kernel_launch(void* const* d_in, const int* in_sizes, int n_in,
                              void* d_out, int out_size, void* d_ws, size_t ws_size,
                              hipStream_t stream)
{
  (void)in_sizes; (void)n_in; (void)out_size; (void)ws_size;

  const float* Q  = (const float*)d_in[0];
  const float* K  = (const float*)d_in[1];
  const float* V  = (const float*)d_in[2];
  /* d_in[3] attn_mask: ignored by the reference forward */
  const unsigned char* em = (const unsigned char*)d_in[4]; // bool edge_mask
  const float* Wq = (const float*)d_in[5];
  const float* bq = (const float*)d_in[6];
  const float* Wk = (const float*)d_in[7];
  const float* bk = (const float*)d_in[8];
  const float* Wv = (const float*)d_in[9];
  const float* bv = (const float*)d_in[10];
  const float* Wo = (const float*)d_in[11];
  const float* bo = (const float*)d_in[12];
  const float* gamma = (const float*)d_in[13];
  const float* beta  = (const float*)d_in[14];

  // Workspace: 5 matrices of [4096,1024] fp32 = 80 MB total.
  float* ws  = (float*)d_ws;
  const size_t MAT = (size_t)M_TOTAL * D_MODEL;
  float* qp  = ws;
  float* kp  = ws + MAT;
  float* vp  = ws + 2 * MAT;
  float* ctx = ws + 3 * MAT;
  float* xb  = ws + 4 * MAT;

  float* outp  = (float*)d_out;      // [B,S,D_MODEL]
  float* attnp = outp + MAT;         // [B,H,S,S] follows in tuple order

  dim3 gg(M_TOTAL / 64, D_MODEL / 32);
  gemm_wmma_f32<<<gg, 256, 0, stream>>>(Q, Wq, bq, nullptr, qp, M_TOTAL, D_MODEL, D_MODEL);
  gemm_wmma_f32<<<gg, 256, 0, stream>>>(K, Wk, bk, nullptr, kp, M_TOTAL, D_MODEL, D_MODEL);
  gemm_wmma_f32<<<gg, 256, 0, stream>>>(V, Wv, bv, nullptr, vp, M_TOTAL, D_MODEL, D_MODEL);

  size_t smem = (size_t)(16 * 64 + 16 * SEQ) * sizeof(float); // 132 KB < 320 KB/WGP
  hipFuncSetAttribute((const void*)attn_wmma,
                      hipFuncAttributeMaxDynamicSharedMemorySize, (int)smem);
  dim3 ga(SEQ / 16, N_HEADS, BATCH);
  attn_wmma<<<ga, 256, smem, stream>>>(qp, kp, vp, em, attnp, ctx);

  gemm_wmma_f32<<<gg, 256, 0, stream>>>(ctx, Wo, bo, Q, xb, M_TOTAL, D_MODEL, D_MODEL);
  layernorm_k<<<M_TOTAL, 256, 0, stream>>>(xb, gamma, beta, outp);
}